// GCN_2800318677548
// MI455X (gfx1250) — compile-verified
//
#include <hip/hip_runtime.h>
#include <hip/hip_bf16.h>

typedef __attribute__((ext_vector_type(2))) float v2f;
typedef __attribute__((ext_vector_type(8))) float v8f;

#define TK 128     // K depth (IN_FEATS == HIDDEN == 128)
#define LDK 132    // padded LDS row stride (floats) to avoid 16-way bank conflicts

// ---------------------------------------------------------------------------
// degree accumulation: degO[src[e]] += 1 ; degI[dst[e]] += 1
// ---------------------------------------------------------------------------
__global__ void gcn_deg_kernel(const int* __restrict__ src, const int* __restrict__ dst,
                               float* __restrict__ degO, float* __restrict__ degI, int E) {
  int t = blockIdx.x * blockDim.x + threadIdx.x;
  if (t < E) {
    atomicAdd(&degO[src[t]], 1.0f);
    atomicAdd(&degI[dst[t]], 1.0f);
  }
}

// in-place deg -> inv_sqrt(deg) (0 if deg == 0), matching the reference
__global__ void gcn_invsqrt_kernel(float* __restrict__ degO, float* __restrict__ degI, int N) {
  int t = blockIdx.x * blockDim.x + threadIdx.x;
  if (t < N) {
    float o = degO[t];
    float i = degI[t];
    degO[t] = (o > 0.0f) ? rsqrtf(fmaxf(o, 1.0f)) : 0.0f;
    degI[t] = (i > 0.0f) ? rsqrtf(fmaxf(i, 1.0f)) : 0.0f;
  }
}

// ---------------------------------------------------------------------------
// fused gather+scale+scatter-add: agg[dst[e]] += h[src[e]] * invO[src[e]]
// one thread per (edge, float4-chunk); CHUNKS is compile-time so the e/c split
// is a shift (CHUNKS=32) or magic-multiply (CHUNKS=10) — no 64-bit division.
// Consecutive threads cover consecutive chunks of one edge -> coalesced
// 16B-per-lane reads and contiguous atomic bursts, all L2-resident.
// ---------------------------------------------------------------------------
template <int CHUNKS>
__global__ void gcn_scatter_kernel(const float* __restrict__ h,
                                   const int* __restrict__ src,
                                   const int* __restrict__ dst,
                                   const float* __restrict__ invO,
                                   float* __restrict__ agg,
                                   int E) {
  constexpr int F = CHUNKS * 4;
  long long t = (long long)blockIdx.x * blockDim.x + threadIdx.x;
  long long total = (long long)E * CHUNKS;
  if (t >= total) return;
  int e = (int)(t / CHUNKS);
  int c = (int)(t - (long long)e * CHUNKS);
  int s = src[e];
  int d = dst[e];
  float sc = invO[s];
  float4 v = ((const float4*)(h + (size_t)s * F))[c];
  float* ad = agg + (size_t)d * F + (size_t)c * 4;
  atomicAdd(ad + 0, v.x * sc);
  atomicAdd(ad + 1, v.y * sc);
  atomicAdd(ad + 2, v.z * sc);
  atomicAdd(ad + 3, v.w * sc);
}

// ---------------------------------------------------------------------------
// C[M x Nd] = act( (A * rowScale) @ W[128 x Nd] + bias )   via V_WMMA_F32_16X16X4_F32
// block = 256 threads (8 wave32), one 16-row M tile per block, one 16-col N tile
// per wave. A tile staged in LDS (row-scaled on load), W streamed (L2 resident).
// Two independent accumulators break the WMMA RAW chain; B loads are
// unconditional (column index clamped; invalid columns never stored).
// ---------------------------------------------------------------------------
__global__ void __launch_bounds__(256)
gcn_gemm_kernel(const float* __restrict__ A,
                const float* __restrict__ rowScale,   // may be null
                const float* __restrict__ W,          // 128 x Nd, row-major
                const float* __restrict__ bias,       // may be null
                float* __restrict__ C,
                int M, int Nd, int relu) {
  __shared__ float As[16 * LDK];

  const int t     = threadIdx.x;
  const int mBase = blockIdx.x * 16;

  // cooperative load of the 16x128 A tile (512 float4s across 256 threads)
  for (int i = t; i < 16 * (TK / 4); i += 256) {
    int row = i >> 5;          // / (TK/4)
    int c4  = i & 31;          // % (TK/4)
    int gr  = mBase + row;
    float4 v = make_float4(0.f, 0.f, 0.f, 0.f);
    float  s = 1.0f;
    if (gr < M) {
      v = ((const float4*)(A + (size_t)gr * TK))[c4];
      if (rowScale) s = rowScale[gr];
    }
    float* dstp = As + (size_t)row * LDK + (size_t)c4 * 4;
    dstp[0] = v.x * s; dstp[1] = v.y * s; dstp[2] = v.z * s; dstp[3] = v.w * s;
  }
  __syncthreads();

  const int wave   = t >> 5;
  const int lane   = t & 31;
  const int nTiles = (Nd + 15) >> 4;
  if (wave >= nTiles) return;

  const int col    = lane & 15;
  const int gcol   = wave * 16 + col;
  const int khalf  = (lane >> 4) * 2;         // lanes 0-15 -> K{0,1}, 16-31 -> K{2,3}
  const bool colOk = (gcol < Nd);
  const int gcSafe = colOk ? gcol : (Nd - 1); // clamp: keeps loads in-bounds,
                                              // garbage only feeds suppressed cols

  const float* As_row = As + (size_t)(lane & 15) * LDK + khalf;
  const float* Wp     = W + (size_t)khalf * Nd + gcSafe;

  v8f acc0 = {};
  v8f acc1 = {};
  #pragma unroll 8
  for (int k0 = 0; k0 < TK; k0 += 8) {
    v2f a0, b0, a1, b1;
    a0[0] = As_row[k0];
    a0[1] = As_row[k0 + 1];
    b0[0] = Wp[0];
    b0[1] = Wp[Nd];
    a1[0] = As_row[k0 + 4];
    a1[1] = As_row[k0 + 5];
    b1[0] = Wp[4 * (size_t)Nd];
    b1[1] = Wp[5 * (size_t)Nd];
    acc0 = __builtin_amdgcn_wmma_f32_16x16x4_f32(false, a0, false, b0, (short)0, acc0,
                                                 false, false);
    acc1 = __builtin_amdgcn_wmma_f32_16x16x4_f32(false, a1, false, b1, (short)0, acc1,
                                                 false, false);
    Wp += 8 * (size_t)Nd;
  }
  v8f acc = acc0 + acc1;

  if (colOk) {
    const float bb    = bias ? bias[gcol] : 0.0f;
    const int   rbase = mBase + (lane >> 4) * 8;   // C/D: VGPR v -> row v + 8*lane[4]
    #pragma unroll
    for (int v = 0; v < 8; ++v) {
      int r = rbase + v;
      if (r < M) {
        float val = acc[v] + bb;
        if (relu) val = fmaxf(val, 0.0f);
        C[(size_t)r * Nd + gcol] = val;
      }
    }
  }
}

// ---------------------------------------------------------------------------
// out[n, c] = agg[n, c] * invI[n] + b3[c]     (final layer, no activation)
// ---------------------------------------------------------------------------
__global__ void gcn_final_kernel(const float* __restrict__ agg,
                                 const float* __restrict__ invI,
                                 const float* __restrict__ b3,
                                 float* __restrict__ out, int N, int Cc) {
  int t = blockIdx.x * blockDim.x + threadIdx.x;
  if (t < N * Cc) {
    int n = t / Cc;
    int c = t - n * Cc;
    out[t] = agg[t] * invI[n] + b3[c];
  }
}

// ---------------------------------------------------------------------------
extern "C" void kernel_launch(void* const* d_in, const int* in_sizes, int n_in,
                              void* d_out, int out_size, void* d_ws, size_t ws_size,
                              hipStream_t stream) {
  const float* feats = (const float*)d_in[0];
  const int*   src   = (const int*)d_in[1];
  const int*   dst   = (const int*)d_in[2];
  const float* W1    = (const float*)d_in[3];
  const float* b1    = (const float*)d_in[4];
  const float* W2    = (const float*)d_in[5];
  const float* b2    = (const float*)d_in[6];
  const float* W3    = (const float*)d_in[7];
  const float* b3    = (const float*)d_in[8];
  float*       out   = (float*)d_out;

  const int HID  = in_sizes[4];              // 128
  const int NCLS = in_sizes[8];              // 40
  const int INF  = in_sizes[3] / HID;        // 128
  const int N    = in_sizes[0] / INF;        // 50000
  const int E    = in_sizes[1];              // 800000

  // workspace layout (floats): invO | invI | bufA (N*128) | bufB (N*128) | bufC (N*NCLS)
  float* invO = (float*)d_ws;
  float* invI = invO + N;
  float* bufA = invI + N;
  float* bufB = bufA + (size_t)N * TK;
  float* bufC = bufB + (size_t)N * TK;

  const int B = 256;
  const int mTiles = (N + 15) / 16;          // 3125

  // --- degrees -> inv_sqrt ---
  hipMemsetAsync(invO, 0, sizeof(float) * 2 * (size_t)N, stream);
  gcn_deg_kernel<<<(E + B - 1) / B, B, 0, stream>>>(src, dst, invO, invI, E);
  gcn_invsqrt_kernel<<<(N + B - 1) / B, B, 0, stream>>>(invO, invI, N);

  // --- layer 1: agg = scatter(feats * invO); h1 = relu((agg * invI) @ W1 + b1) ---
  hipMemsetAsync(bufA, 0, sizeof(float) * (size_t)N * TK, stream);
  {
    long long tot = (long long)E * 32;
    gcn_scatter_kernel<32><<<(int)((tot + B - 1) / B), B, 0, stream>>>(feats, src, dst, invO, bufA, E);
  }
  gcn_gemm_kernel<<<mTiles, B, 0, stream>>>(bufA, invI, W1, b1, bufB, N, HID, 1);

  // --- layer 2 ---
  hipMemsetAsync(bufA, 0, sizeof(float) * (size_t)N * TK, stream);
  {
    long long tot = (long long)E * 32;
    gcn_scatter_kernel<32><<<(int)((tot + B - 1) / B), B, 0, stream>>>(bufB, src, dst, invO, bufA, E);
  }
  gcn_gemm_kernel<<<mTiles, B, 0, stream>>>(bufA, invI, W2, b2, bufB, N, HID, 1);

  // --- layer 3: project first (40 < 128), then message-pass 40 feats ---
  gcn_gemm_kernel<<<mTiles, B, 0, stream>>>(bufB, nullptr, W3, nullptr, bufC, N, NCLS, 0);
  hipMemsetAsync(bufA, 0, sizeof(float) * (size_t)N * NCLS, stream);
  {
    long long tot = (long long)E * 10;
    gcn_scatter_kernel<10><<<(int)((tot + B - 1) / B), B, 0, stream>>>(bufC, src, dst, invO, bufA, E);
  }
  gcn_final_kernel<<<(N * NCLS + B - 1) / B, B, 0, stream>>>(bufA, invI, b3, out, N, NCLS);
}